// TopKRouter_38302518346149
// MI455X (gfx1250) — compile-verified
//
#include <hip/hip_runtime.h>

typedef float v2f __attribute__((ext_vector_type(2)));
typedef float v8f __attribute__((ext_vector_type(8)));

namespace {
constexpr int D_DIM  = 4096;   // hidden dim
constexpr int N_EXP  = 64;     // experts
constexpr int KC     = 128;    // K-chunk staged in LDS
constexpr int WPITCH = KC + 4; // 132: bank stride 4 (conflict-free), 16B-aligned rows
constexpr int ROWS   = 128;    // tokens per block
constexpr int TOKENS = 4 * 4096;
constexpr int OPITCH = 65;     // padded pitch for phase-2 logits tile
constexpr int NCHUNK = D_DIM / KC;            // 32
constexpr unsigned BUF_FLOATS = N_EXP * WPITCH;      // 8448
constexpr unsigned BUF_BYTES  = BUF_FLOATS * 4;      // 33792
}

// Issue one W K-chunk (64 x 128 f32 = 32 KB) as async global->LDS copies.
// 256 threads x 8 iters x 16 B/lane. No VGPR round-trip; tracked by ASYNCcnt.
__device__ __forceinline__ void stage_w_async(const float* __restrict__ Wc,
                                              int tid, unsigned ldsBase)
{
    #pragma unroll
    for (int i = 0; i < 8; ++i) {
        const int f  = tid + i * 256;          // 0..2047 float4 slots
        const int e  = f >> 5;                 // 32 float4 per expert row
        const int c4 = f & 31;
        const unsigned goff = (unsigned)(e * D_DIM + c4 * 4) * 4u;           // bytes
        const unsigned loff = ldsBase + (unsigned)(e * WPITCH + c4 * 4) * 4u; // bytes
        asm volatile("global_load_async_to_lds_b128 %0, %1, %2"
                     :
                     : "v"(loff), "v"(goff), "s"(Wc)
                     : "memory");
    }
}

__device__ __forceinline__ void wait_async_zero()
{
    asm volatile("s_wait_asynccnt 0x0" ::: "memory");
}

__global__ __launch_bounds__(256)
void router_gemm_top2_kernel(const float* __restrict__ x,
                             const float* __restrict__ W,
                             float* __restrict__ probs,
                             int* __restrict__ idx)
{
    // Double-buffered W chunks; buffer 0 reused as [128][65] logits in phase 2.
    __shared__ float lds[2 * BUF_FLOATS];

    const int tid  = threadIdx.x;
    const int wave = tid >> 5;
    const int lane = tid & 31;
    const int l16  = lane & 15;
    const int half = lane >> 4;          // 0: K pair {0,1}; 1: K pair {2,3}

    const int tokBase = blockIdx.x * ROWS;
    // WMMA f32 A fragment: lanes 0-15 and 16-31 both map M = lane%16
    const float* __restrict__ xrow =
        x + (size_t)(tokBase + wave * 16 + l16) * D_DIM;

    v8f acc0 = {0.f,0.f,0.f,0.f,0.f,0.f,0.f,0.f};
    v8f acc1 = {0.f,0.f,0.f,0.f,0.f,0.f,0.f,0.f};
    v8f acc2 = {0.f,0.f,0.f,0.f,0.f,0.f,0.f,0.f};
    v8f acc3 = {0.f,0.f,0.f,0.f,0.f,0.f,0.f,0.f};

    // Prologue: async-stage chunk 0 into buffer 0.
    stage_w_async(W, tid, 0u);
    wait_async_zero();
    __syncthreads();

    int buf = 0;
    for (int chunk = 0; chunk < NCHUNK; ++chunk) {
        const int kc = chunk * KC;

        // Prefetch next chunk into this same buffer (buf) while computing this one.
        if (chunk + 1 < NCHUNK)
            stage_w_async(W + (kc + KC), tid, (buf ^ 1) ? 0u : BUF_BYTES);

        const float* __restrict__ wbuf = lds + (unsigned)buf * BUF_FLOATS;

        #pragma unroll 4
        for (int j = 0; j < KC; j += 4) {
            const int kk = j + half * 2;

            // A fragment: row = token, cols kk, kk+1 (per-half K pairs)
            v2f a;
            a.x = xrow[kc + kk];
            a.y = xrow[kc + kk + 1];

            // B fragments: lane -> expert column, half -> K pair, from LDS
            const float* wb = wbuf + kk;
            v2f b0, b1, b2, b3;
            b0.x = wb[( 0 + l16) * WPITCH]; b0.y = wb[( 0 + l16) * WPITCH + 1];
            b1.x = wb[(16 + l16) * WPITCH]; b1.y = wb[(16 + l16) * WPITCH + 1];
            b2.x = wb[(32 + l16) * WPITCH]; b2.y = wb[(32 + l16) * WPITCH + 1];
            b3.x = wb[(48 + l16) * WPITCH]; b3.y = wb[(48 + l16) * WPITCH + 1];

            acc0 = __builtin_amdgcn_wmma_f32_16x16x4_f32(false, a, false, b0,
                                                         (short)0, acc0, false, false);
            acc1 = __builtin_amdgcn_wmma_f32_16x16x4_f32(false, a, false, b1,
                                                         (short)0, acc1, false, false);
            acc2 = __builtin_amdgcn_wmma_f32_16x16x4_f32(false, a, false, b2,
                                                         (short)0, acc2, false, false);
            acc3 = __builtin_amdgcn_wmma_f32_16x16x4_f32(false, a, false, b3,
                                                         (short)0, acc3, false, false);
        }

        // My prefetch landed in LDS; barrier makes it visible to all waves and
        // confirms everyone is done reading the buffer we overwrite next.
        wait_async_zero();
        __syncthreads();
        buf ^= 1;
    }

    // Phase 2: scatter 16x64 logits tile per wave to LDS (buffer 0), token-major.
    // C layout: VGPR r holds M=r (lanes 0-15) / M=r+8 (lanes 16-31), N = lane%16.
    #pragma unroll
    for (int r = 0; r < 8; ++r) {
        const int tl = wave * 16 + half * 8 + r;
        lds[tl * OPITCH +  0 + l16] = acc0[r];
        lds[tl * OPITCH + 16 + l16] = acc1[r];
        lds[tl * OPITCH + 32 + l16] = acc2[r];
        lds[tl * OPITCH + 48 + l16] = acc3[r];
    }
    __syncthreads();

    // One thread per token: top-2 scan (stable, lowest index wins ties) + softmax.
    if (tid < ROWS) {
        const float* rowv = lds + tid * OPITCH;
        float s1 = -3.402823466e38f, s2 = -3.402823466e38f;
        int i1 = 0, i2 = 0;
        #pragma unroll
        for (int e = 0; e < N_EXP; ++e) {
            const float v = rowv[e];
            if (v > s1)      { s2 = s1; i2 = i1; s1 = v; i1 = e; }
            else if (v > s2) { s2 = v; i2 = e; }
        }
        const float p1  = __expf(s2 - s1);   // stable 2-way softmax
        const float inv = 1.0f / (1.0f + p1);
        const int tok = tokBase + tid;
        probs[tok * 2 + 0] = inv;
        probs[tok * 2 + 1] = p1 * inv;
        idx[tok * 2 + 0] = i1;
        idx[tok * 2 + 1] = i2;
    }
}

extern "C" void kernel_launch(void* const* d_in, const int* in_sizes, int n_in,
                              void* d_out, int out_size, void* d_ws, size_t ws_size,
                              hipStream_t stream) {
    const float* x = (const float*)d_in[0];   // [4,4096,4096] f32
    const float* W = (const float*)d_in[1];   // [64,4096]     f32

    float* probs = (float*)d_out;                 // [16384,2] f32
    int*   idx   = (int*)d_out + TOKENS * 2;      // [16384,2] i32 (same 4B stride)

    dim3 grid(TOKENS / ROWS);   // 128 blocks
    dim3 block(256);            // 8 wave32s
    router_gemm_top2_kernel<<<grid, block, 0, stream>>>(x, W, probs, idx);
}